// PonderIndoor_44186623541472
// MI455X (gfx1250) — compile-verified
//
#include <hip/hip_runtime.h>

// ---------------------------------------------------------------------------
// PonderIndoor scatter-mean voxelization for MI455X (gfx1250, wave32).
//
// Strategy (memory-bound problem, ~1.2 GB mandatory traffic -> ~52us floor):
//   * Loop over the 8 batches; one batch's [64^3, 96] f32 accumulator is
//     ~100.7 MB and stays resident in the 192 MB L2 across the
//     zero -> scatter -> transform sequence.
//   * Scatter: one wave32 per point; 3 coalesced global_atomic_add_f32 per
//     lane cover the 96 contiguous channels of the point's segment.
//   * Transform: divide by count and permute [x][y][z][c] -> [c][z][y][x].
//     The 16x16 (x<->c) lane transpose is done with V_WMMA_F32_16X16X4_F32
//     using shifted-identity B matrices (exact in f32), so both the
//     accumulator read and the 805 MB output write are fully coalesced.
// ---------------------------------------------------------------------------

typedef __attribute__((ext_vector_type(2))) float v2f;
typedef __attribute__((ext_vector_type(8))) float v8f;

#define GRID   64
#define G3     (GRID * GRID * GRID)
#define CHN    96
#define VOXSZ  0.02f

// ---------------------------------------------------------------------------
// Pass 1: wave-per-point scatter into per-batch accumulator (L2 resident).
// sums layout: [seg][96] contiguous; cnts: [seg].
// ---------------------------------------------------------------------------
__global__ void __launch_bounds__(256)
scatter_kernel(const float* __restrict__ coord,
               const float* __restrict__ feat,
               const int*   __restrict__ offset,
               const int*   __restrict__ resolution,
               int b, int npb,
               float* __restrict__ sums,
               float* __restrict__ cnts) {
    int gtid = blockIdx.x * blockDim.x + threadIdx.x;
    int wave = gtid >> 5;        // one wave32 per point
    int lane = gtid & 31;

    int start = (b == 0) ? 0 : offset[b - 1];
    int end   = offset[b];
    int p     = start + wave;
    if (wave >= npb || p >= end) return;   // wave-uniform exit

    // voxelization, matching reference float math:
    //   vox  = floor(coord / 0.02); gidx = floor(vox / ((res+1)/64)); clip
    float cx = coord[3 * (size_t)p + 0];
    float cy = coord[3 * (size_t)p + 1];
    float cz = coord[3 * (size_t)p + 2];
    float s  = (float)(resolution[b] + 1) / (float)GRID;

    int gx = (int)floorf(floorf(cx / VOXSZ) / s);
    int gy = (int)floorf(floorf(cy / VOXSZ) / s);
    int gz = (int)floorf(floorf(cz / VOXSZ) / s);
    gx = min(max(gx, 0), GRID - 1);
    gy = min(max(gy, 0), GRID - 1);
    gz = min(max(gz, 0), GRID - 1);
    int seg = (gx * GRID + gy) * GRID + gz;

    const float* f   = feat + (size_t)p * CHN;
    float*       dst = sums + (size_t)seg * CHN;
    // 96 channels / 32 lanes = 3 coalesced atomic adds (384 contiguous bytes)
    atomicAdd(dst + lane,      f[lane]);
    atomicAdd(dst + lane + 32, f[lane + 32]);
    atomicAdd(dst + lane + 64, f[lane + 64]);
    if (lane == 0) atomicAdd(cnts + seg, 1.0f);
}

// ---------------------------------------------------------------------------
// Pass 2: mean + permute. One wave32 per 16(x) x 16(c) tile at fixed (y,z).
//
// WMMA identity-transpose trick:
//   Load A[m][k] = S[x = 4j+k][c = m]  (lane index carries c -> coalesced
//   row reads of the [seg][96] accumulator; the K half-split across lane
//   halves matches the ISA 16x4 f32 A layout).
//   B_j[k][n] = delta(n, 4j+k) (shifted identity, built from lane compares).
//   After 4 accumulating WMMAs: D[m][n] = S[x=n][c=m] in the D layout
//   (VGPR r = row m, lanes carry n) -> lanes now carry x: stores to the
//   [c][z][y][x] output are fully coalesced.
// ---------------------------------------------------------------------------
__global__ void __launch_bounds__(256)
transform_kernel(const float* __restrict__ sums,
                 const float* __restrict__ cnts,
                 float* __restrict__ outb) {
    int gtid = blockIdx.x * blockDim.x + threadIdx.x;
    int wave = gtid >> 5;
    int lane = gtid & 31;

    // tile decomposition: wave = ((y*64 + z)*6 + ct)*4 + xt
    int xt = wave & 3;
    int ct = (wave >> 2) % 6;
    int yz = wave / 24;                 // 0..4095 (exact grid)
    int z  = yz & (GRID - 1);
    int y  = yz >> 6;

    int n  = lane & 15;                 // within-tile x after transpose / c before
    int h  = lane >> 4;                 // lane half
    int x0 = xt * 16;
    int c0 = ct * 16;

    // per-lane reciprocal count for x = x0 + n (1 MB cnt array, L2 hit)
    int   segx = ((x0 + n) * GRID + y) * GRID + z;
    float rc   = 1.0f / fmaxf(cnts[segx], 1.0f);

    const size_t xstride = (size_t)GRID * GRID * CHN;  // +1 in x
    v8f acc = {};
#pragma unroll
    for (int j = 0; j < 4; ++j) {
        int xr = 4 * j + 2 * h;         // K rows covered by this lane half
        size_t s0 = ((size_t)((x0 + xr) * GRID + y) * GRID + z) * CHN + c0 + n;
        v2f a;                          // A_j: VGPR0 = k even half, VGPR1 = odd
        a.x = sums[s0];
        a.y = sums[s0 + xstride];
        v2f bv;                         // B_j shifted identity in B layout
        bv.x = (n == 4 * j + 2 * h)     ? 1.0f : 0.0f;
        bv.y = (n == 4 * j + 2 * h + 1) ? 1.0f : 0.0f;
        acc = __builtin_amdgcn_wmma_f32_16x16x4_f32(
            /*neg_a=*/false, a, /*neg_b=*/false, bv,
            /*c_mod=*/(short)0, acc, /*reuse_a=*/false, /*reuse_b=*/false);
    }

    // acc[r]: lanes 0-15 -> row c = c0+r, lanes 16-31 -> c = c0+r+8; lane = x
#pragma unroll
    for (int r = 0; r < 8; ++r) {
        size_t o = (((size_t)(c0 + r + 8 * h) * GRID + z) * GRID + y) * GRID
                   + (x0 + n);
        outb[o] = acc[r] * rc;
    }
}

// ---------------------------------------------------------------------------
extern "C" void kernel_launch(void* const* d_in, const int* in_sizes, int n_in,
                              void* d_out, int out_size, void* d_ws, size_t ws_size,
                              hipStream_t stream) {
    (void)n_in; (void)out_size; (void)ws_size;
    const float* coord      = (const float*)d_in[0];   // [N,3]
    const float* feat       = (const float*)d_in[1];   // [N,96]
    const int*   offset     = (const int*)d_in[2];     // [B] cumulative ends
    const int*   resolution = (const int*)d_in[3];     // [B]
    float*       out        = (float*)d_out;           // [B,96,64,64,64]

    int B    = in_sizes[2];
    int npts = in_sizes[0] / 3;
    int npb  = npts / B;

    float* sums = (float*)d_ws;                        // [64^3, 96]  ~100.7 MB
    float* cnts = sums + (size_t)G3 * CHN;             // [64^3]      ~1 MB
    size_t zero_bytes = ((size_t)G3 * CHN + (size_t)G3) * sizeof(float);

    int scatter_blocks   = (npb * 32 + 255) / 256;               // wave per point
    int transform_blocks = (GRID * GRID * 6 * 4) * 32 / 256;     // wave per tile

    for (int b = 0; b < B; ++b) {
        hipMemsetAsync(d_ws, 0, zero_bytes, stream);
        scatter_kernel<<<scatter_blocks, 256, 0, stream>>>(
            coord, feat, offset, resolution, b, npb, sums, cnts);
        transform_kernel<<<transform_blocks, 256, 0, stream>>>(
            sums, cnts, out + (size_t)b * CHN * G3);
    }
}